// DBScan_38585986187994
// MI455X (gfx1250) — compile-verified
//
#include <hip/hip_runtime.h>
#include <hip/hip_bf16.h>

typedef __attribute__((ext_vector_type(2))) float v2f;
typedef __attribute__((ext_vector_type(8))) float v8f;

#define NPTS   16384
#define NTILES 1024            // NPTS / 16
#define NWORDS 512             // NPTS / 32
#define EPS2   0.04f           // 0.2^2
#define MINPTS 10u
#define BIGLAB 0x7FFFFFFFu

__device__ inline unsigned wave_min(unsigned m) {
#pragma unroll
    for (int off = 16; off > 0; off >>= 1)
        m = min(m, (unsigned)__shfl_xor((int)m, off, 32));
    return m;
}

__device__ inline unsigned wave_sum(unsigned v) {
#pragma unroll
    for (int off = 16; off > 0; off >>= 1)
        v += (unsigned)__shfl_xor((int)v, off, 32);
    return v;
}

// ---------------------------------------------------------------------------
// Phase 1: pairwise d^2 via one V_WMMA_F32_16X16X4_F32 per 16x16 tile.
//   A row i = [x, y, z, |p|^2], B col j = [-2x, -2y, -2z, 1], C = |p_j|^2
//   => D = full squared-distance tile.
// Bit packing uses SYMMETRY: each lane packs its COLUMN's 16 bits (one
// shfl_xor(16) merges the row-0..7 / row-8..15 lane halves) and stores them
// at adj16[col][rowTile] — which, because adj is symmetric, is exactly the
// row-major bitmask adj16[row][tile] once all tiles are written. No ballot,
// no transpose, no divergent selects in the hot loop.
// ---------------------------------------------------------------------------
__global__ __launch_bounds__(256)
void adj_kernel(const float* __restrict__ pts,
                unsigned short* __restrict__ adj16) {
    const int lane    = threadIdx.x & 31;
    const int wave    = threadIdx.x >> 5;
    const int rowTile = blockIdx.x * 8 + wave;
    const int i0      = rowTile * 16;
    const int l       = lane & 15;

    // --- A operand (16x4 f32): lanes 0-15 hold K=0,1 ; lanes 16-31 hold K=2,3
    const int m  = i0 + l;
    const float px = pts[3 * m + 0];
    const float py = pts[3 * m + 1];
    const float pz = pts[3 * m + 2];
    const float sqi = px * px + py * py + pz * pz;
    v2f a;
    a.x = (lane < 16) ? px : pz;     // K=0 | K=2
    a.y = (lane < 16) ? py : sqi;    // K=1 | K=3

    for (int j = 0; j < NTILES; ++j) {
        const int cidx = j * 16 + l;
        const float qx = pts[3 * cidx + 0];
        const float qy = pts[3 * cidx + 1];
        const float qz = pts[3 * cidx + 2];
        const float sqj = qx * qx + qy * qy + qz * qz;

        if (j + 1 < NTILES)  // pull next tile's points into near caches
            __builtin_prefetch(&pts[3 * ((j + 1) * 16 + l)], 0, 3);

        // --- B operand (4x16 f32): lanes 0-15 hold K=0,1 ; lanes 16-31 K=2,3
        v2f b;
        b.x = (lane < 16) ? -2.0f * qx : -2.0f * qz;  // K=0 | K=2
        b.y = (lane < 16) ? -2.0f * qy : 1.0f;        // K=1 | K=3

        // --- C operand: |p_j|^2 broadcast down each column
        v8f c;
#pragma unroll
        for (int r = 0; r < 8; ++r) c[r] = sqj;

        v8f d = __builtin_amdgcn_wmma_f32_16x16x4_f32(
            false, a, false, b, (short)0, c, false, false);

        // D layout: lane L<16 reg r -> (row r,  col L)
        //           lane L>=16 reg r -> (row r+8, col L-16)
        unsigned mask8 = 0u;
#pragma unroll
        for (int r = 0; r < 8; ++r)
            mask8 |= (d[r] < EPS2 ? 1u : 0u) << r;

        // merge halves: lanes 0-15 get rows 8-15 from lanes 16-31
        const unsigned other = (unsigned)__shfl_xor((int)mask8, 16, 32);
        const unsigned colmask = mask8 | (other << 8);  // valid in lanes 0-15

        if (lane < 16)  // symmetric store: column mask == row mask
            adj16[(size_t)(j * 16 + lane) * NTILES + rowTile] =
                (unsigned short)colmask;
    }
}

// ---------------------------------------------------------------------------
// Phase 2: degree (popcount sweep of the L2-resident bitmask, wave per row)
//          fused with core flags + initial labels.
// ---------------------------------------------------------------------------
__global__ __launch_bounds__(256)
void init_kernel(const unsigned* __restrict__ adj32,
                 unsigned* __restrict__ core,
                 unsigned* __restrict__ lab) {
    const int lane = threadIdx.x & 31;
    const int row  = (int)((blockIdx.x * blockDim.x + threadIdx.x) >> 5);

    unsigned cnt = 0;
    const unsigned* rowp = adj32 + (size_t)row * NWORDS;
    for (int w = lane; w < NWORDS; w += 32) cnt += __popc(rowp[w]);
    cnt = wave_sum(cnt);

    if (lane == 0) {
        const unsigned c = (cnt >= MINPTS) ? 1u : 0u;
        core[row] = c;
        lab[row]  = c ? (unsigned)row : BIGLAB;
    }
}

// ---------------------------------------------------------------------------
// Phase 3: in-place min-label propagation over core<->core edges.
// One wave scans one adjacency row (512 u32 words, L2-resident).
// ---------------------------------------------------------------------------
__global__ __launch_bounds__(256)
void prop_kernel(const unsigned* __restrict__ adj32,
                 const unsigned* __restrict__ core,
                 unsigned* __restrict__ lab) {
    const int lane = threadIdx.x & 31;
    const int row  = (int)((blockIdx.x * blockDim.x + threadIdx.x) >> 5);
    if (!core[row]) return;  // wave-uniform

    unsigned m = BIGLAB;
    const unsigned* rowp = adj32 + (size_t)row * NWORDS;
    for (int w = lane; w < NWORDS; w += 32) {
        unsigned bits = rowp[w];
        while (bits) {
            const int b = __builtin_ctz(bits);
            bits &= bits - 1;
            const int j = w * 32 + b;
            if (core[j]) m = min(m, lab[j]);
        }
    }
    m = wave_min(m);
    if (lane == 0 && m < BIGLAB) atomicMin(&lab[row], m);
}

// ---------------------------------------------------------------------------
// Phase 4: every point's root = min label among its CORE neighbors.
// ---------------------------------------------------------------------------
__global__ __launch_bounds__(256)
void root_kernel(const unsigned* __restrict__ adj32,
                 const unsigned* __restrict__ core,
                 const unsigned* __restrict__ lab,
                 unsigned* __restrict__ root) {
    const int lane = threadIdx.x & 31;
    const int row  = (int)((blockIdx.x * blockDim.x + threadIdx.x) >> 5);

    unsigned m = BIGLAB;
    const unsigned* rowp = adj32 + (size_t)row * NWORDS;
    for (int w = lane; w < NWORDS; w += 32) {
        unsigned bits = rowp[w];
        while (bits) {
            const int b = __builtin_ctz(bits);
            bits &= bits - 1;
            const int j = w * 32 + b;
            if (core[j]) m = min(m, lab[j]);
        }
    }
    m = wave_min(m);
    if (lane == 0) root[row] = m;
}

// ---------------------------------------------------------------------------
// Phase 5: rank[i] = cumsum(is_root)[i] - 1   (single 1024-thread block)
// ---------------------------------------------------------------------------
__global__ __launch_bounds__(1024)
void scan_kernel(const unsigned* __restrict__ core,
                 const unsigned* __restrict__ lab,
                 int* __restrict__ rank) {
    __shared__ int sums[1024];
    const int t = threadIdx.x;
    const int base = t * 16;
    int cnt = 0;
    int loc[16];
#pragma unroll
    for (int k = 0; k < 16; ++k) {
        const int i = base + k;
        cnt += (core[i] && lab[i] == (unsigned)i) ? 1 : 0;
        loc[k] = cnt;
    }
    sums[t] = cnt;
    __syncthreads();
    for (int off = 1; off < 1024; off <<= 1) {
        const int add = (t >= off) ? sums[t - off] : 0;
        const int v   = sums[t];
        __syncthreads();
        sums[t] = v + add;
        __syncthreads();
    }
    const int excl = (t > 0) ? sums[t - 1] : 0;
#pragma unroll
    for (int k = 0; k < 16; ++k) rank[base + k] = excl + loc[k] - 1;
}

// ---------------------------------------------------------------------------
// Phase 6: labels = reached ? rank[root] : -1   (float32 output)
// ---------------------------------------------------------------------------
__global__ void final_kernel(const unsigned* __restrict__ root,
                             const int* __restrict__ rank,
                             float* __restrict__ out) {
    const int i = blockIdx.x * blockDim.x + threadIdx.x;
    const unsigned r = root[i];
    out[i] = (r < BIGLAB) ? (float)rank[r] : -1.0f;
}

extern "C" void kernel_launch(void* const* d_in, const int* in_sizes, int n_in,
                              void* d_out, int out_size, void* d_ws, size_t ws_size,
                              hipStream_t stream) {
    const float* pts = (const float*)d_in[0];
    float* out = (float*)d_out;

    // workspace layout
    char* p = (char*)d_ws;
    unsigned short* adj16 = (unsigned short*)p;                // 32 MB bitmask
    unsigned*       adj32 = (unsigned*)p;
    p += (size_t)NPTS * NTILES * sizeof(unsigned short);
    unsigned* core = (unsigned*)p; p += NPTS * sizeof(unsigned);
    unsigned* lab  = (unsigned*)p; p += NPTS * sizeof(unsigned);
    unsigned* root = (unsigned*)p; p += NPTS * sizeof(unsigned);
    int*      rank = (int*)p;

    // 1) adjacency bitmask: 1024 row strips, 8 waves/block
    adj_kernel<<<NTILES / 8, 256, 0, stream>>>(pts, adj16);

    // 2) degree + core flags + label init (wave per row)
    init_kernel<<<(NPTS * 32) / 256, 256, 0, stream>>>(adj32, core, lab);

    // 3) fixed-count async min propagation (L2-resident bitmask sweeps)
    for (int it = 0; it < 80; ++it)
        prop_kernel<<<(NPTS * 32) / 256, 256, 0, stream>>>(adj32, core, lab);

    // 4) roots for all points (core + border)
    root_kernel<<<(NPTS * 32) / 256, 256, 0, stream>>>(adj32, core, lab, root);

    // 5) sequential cluster renumbering
    scan_kernel<<<1, 1024, 0, stream>>>(core, lab, rank);

    // 6) final float labels
    final_kernel<<<NPTS / 256, 256, 0, stream>>>(root, rank, out);
}